// PytorchMeanshift_44083544326453
// MI455X (gfx1250) — compile-verified
//
#include <hip/hip_runtime.h>
#include <hip/hip_bf16.h>

// -----------------------------------------------------------------------------
// PytorchMeanshift on gfx1250 (MI455X).
//
// Phase A (tiny, compute-once): bandwidth softmax weights w[4][N][3].
//   logits depend only on X_fea + per-iter weights, NOT on evolving X.
//   K_zero -> K_stats (h = Xfea@W1^T+b1, accumulate per-channel sum/sumsq)
//          -> K_weights (BN with batch stats, ReLU, @W2^T+b2, softmax)
//
// Phase B (dominant, 4x): mean-shift over 10^8 pairs.
//   dist = xn_i + xn_j - 2 * X@X^T ; Gram tiles via V_WMMA_F32_16X16X4_F32
//   (K=3 padded to 4). Flat-kernel masked sums + counts for 3 nested
//   bandwidths in-register (cmp/cndmask/dual-fmac), reduced with shfl_xor +
//   LDS across the block's 8 waves. X (120KB) stays hot in the 192MB L2.
//   v2: software-pipelined X[j] loads (prefetch tile jt+8 while computing on
//   tile jt) so L0/L2 latency overlaps the WMMA + FMA stream.
// -----------------------------------------------------------------------------

typedef __attribute__((ext_vector_type(2))) float v2f;
typedef __attribute__((ext_vector_type(8))) float v8f;

#define NPTS   10000
#define DFEA   32
#define NITER  4
#define NBW    3
#define NTILE  625        // NPTS / 16, exact
#define BN_EPS 1e-5f

// ---------------------------------------------------------------- Phase A ----

__global__ __launch_bounds__(256) void zero_stats_kernel(float* __restrict__ sums) {
    sums[threadIdx.x] = 0.0f;   // 256 floats: [4][32] sums then [4][32] sumsqs
}

__global__ __launch_bounds__(256) void stats_kernel(
        const float* __restrict__ Xfea, const float* __restrict__ W1,
        const float* __restrict__ b1, float* __restrict__ sums)
{
    __shared__ float ls[2 * DFEA];
    const int it = blockIdx.y;
    const int p  = blockIdx.x * 256 + threadIdx.x;
    if (threadIdx.x < 2 * DFEA) ls[threadIdx.x] = 0.0f;
    __syncthreads();
    if (p < NPTS) {
        float f[DFEA];
        #pragma unroll
        for (int k = 0; k < DFEA; ++k) f[k] = Xfea[p * DFEA + k];
        #pragma unroll 4
        for (int c = 0; c < DFEA; ++c) {
            float acc = b1[it * DFEA + c];
            const float* wrow = &W1[(it * DFEA + c) * DFEA];
            #pragma unroll
            for (int k = 0; k < DFEA; ++k) acc = fmaf(f[k], wrow[k], acc);
            atomicAdd(&ls[c], acc);
            atomicAdd(&ls[DFEA + c], acc * acc);
        }
    }
    __syncthreads();
    if (threadIdx.x < 2 * DFEA) {
        const int c     = threadIdx.x & (DFEA - 1);
        const int which = threadIdx.x >> 5;              // 0=sum, 1=sumsq
        atomicAdd(&sums[which * NITER * DFEA + it * DFEA + c], ls[threadIdx.x]);
    }
}

__global__ __launch_bounds__(256) void weights_kernel(
        const float* __restrict__ Xfea, const float* __restrict__ W1,
        const float* __restrict__ b1, const float* __restrict__ gamma,
        const float* __restrict__ beta, const float* __restrict__ W2,
        const float* __restrict__ b2, const float* __restrict__ sums,
        float* __restrict__ wbuf)
{
    const int it = blockIdx.y;
    const int p  = blockIdx.x * 256 + threadIdx.x;
    if (p >= NPTS) return;

    float f[DFEA];
    #pragma unroll
    for (int k = 0; k < DFEA; ++k) f[k] = Xfea[p * DFEA + k];

    float l0 = b2[it * NBW + 0], l1 = b2[it * NBW + 1], l2 = b2[it * NBW + 2];
    const float invN = 1.0f / (float)NPTS;
    #pragma unroll 4
    for (int c = 0; c < DFEA; ++c) {
        float acc = b1[it * DFEA + c];
        const float* wrow = &W1[(it * DFEA + c) * DFEA];
        #pragma unroll
        for (int k = 0; k < DFEA; ++k) acc = fmaf(f[k], wrow[k], acc);
        const float mu  = sums[it * DFEA + c] * invN;
        const float var = sums[NITER * DFEA + it * DFEA + c] * invN - mu * mu;  // biased
        float hn = (acc - mu) * rsqrtf(var + BN_EPS) * gamma[it * DFEA + c]
                 + beta[it * DFEA + c];
        hn = fmaxf(hn, 0.0f);
        l0 = fmaf(hn, W2[(it * NBW + 0) * DFEA + c], l0);
        l1 = fmaf(hn, W2[(it * NBW + 1) * DFEA + c], l1);
        l2 = fmaf(hn, W2[(it * NBW + 2) * DFEA + c], l2);
    }
    const float m  = fmaxf(l0, fmaxf(l1, l2));
    const float e0 = expf(l0 - m), e1 = expf(l1 - m), e2 = expf(l2 - m);
    const float inv = 1.0f / (e0 + e1 + e2);
    float* wr = &wbuf[(size_t)(it * NPTS + p) * NBW];
    wr[0] = e0 * inv; wr[1] = e1 * inv; wr[2] = e2 * inv;
}

// ---------------------------------------------------------------- Phase B ----
// One block = one 16-row tile. 8 waves split the 625 column tiles.
// Gram tile D-layout: lane L, VGPR v -> (M = v + 8*(L>>4), N = L&15).

__global__ __launch_bounds__(256) void meanshift_kernel(
        const float* __restrict__ Xin, const float* __restrict__ w,
        float* __restrict__ Xout)
{
    const int i0   = blockIdx.x * 16;
    const int tid  = threadIdx.x;
    const int wave = tid >> 5;
    const int lane = tid & 31;
    const int half = lane >> 4;
    const int lp   = lane & 15;

    // A operand (16x4 f32): lane = M = lp; VGPR0: K=0 (half 0) / K=2 (half 1),
    // VGPR1: K=1 / K=3. Row vector = (x, y, z, 0).
    const float ax = Xin[3 * (i0 + lp) + 0];
    const float ay = Xin[3 * (i0 + lp) + 1];
    const float az = Xin[3 * (i0 + lp) + 2];
    v2f A;
    A.x = half ? az : ax;
    A.y = half ? 0.0f : ay;

    // |X_i|^2 for the 8 rows this lane owns in the Gram D-layout.
    float dbase[8];
    #pragma unroll
    for (int v = 0; v < 8; ++v) {
        const int r = i0 + v + 8 * half;
        const float x = Xin[3 * r + 0], y = Xin[3 * r + 1], z = Xin[3 * r + 2];
        dbase[v] = x * x + y * y + z * z;
    }

    const float bwsq[NBW] = {0.04f, 2.89f, 10.24f};

    // acc[q][v]: q = bandwidth*4 + {sumx,sumy,sumz,count}
    float acc[4 * NBW][8];
    #pragma unroll
    for (int q = 0; q < 4 * NBW; ++q)
        #pragma unroll
        for (int v = 0; v < 8; ++v) acc[q][v] = 0.0f;

    // ---- software-pipelined column loop (prefetch depth 1) ----
    // All control is wave-uniform: EXEC stays all-ones for every WMMA.
    int jt = wave;
    float xj = Xin[3 * (jt * 16 + lp) + 0];
    float yj = Xin[3 * (jt * 16 + lp) + 1];
    float zj = Xin[3 * (jt * 16 + lp) + 2];

    while (jt < NTILE) {
        // Prefetch next tile's coordinates (clamped index on the last trip:
        // redundant but branch-free and wave-uniform).
        const int jn  = jt + 8;
        const int jp  = (jn < NTILE) ? jn : jt;
        const int j2  = jp * 16 + lp;
        const float nx = Xin[3 * j2 + 0];
        const float ny = Xin[3 * j2 + 1];
        const float nz = Xin[3 * j2 + 2];

        // ---- consume current tile ----
        const float xnj = xj * xj + yj * yj + zj * zj;

        // B operand (4x16 f32): lane = N; K split across lane halves like A.
        v2f B;
        B.x = half ? zj : xj;
        B.y = half ? 0.0f : yj;

        v8f C = {0.0f, 0.0f, 0.0f, 0.0f, 0.0f, 0.0f, 0.0f, 0.0f};
        C = __builtin_amdgcn_wmma_f32_16x16x4_f32(
                /*neg_a=*/false, A, /*neg_b=*/false, B,
                /*c_mod=*/(short)0, C, /*reuse_a=*/false, /*reuse_b=*/false);

        #pragma unroll
        for (int v = 0; v < 8; ++v) {
            const float d = fmaf(C[v], -2.0f, dbase[v] + xnj);   // squared distance
            #pragma unroll
            for (int b = 0; b < NBW; ++b) {
                const float t = (d <= bwsq[b]) ? 1.0f : 0.0f;    // flat kernel
                acc[4 * b + 0][v] = fmaf(t, xj, acc[4 * b + 0][v]);
                acc[4 * b + 1][v] = fmaf(t, yj, acc[4 * b + 1][v]);
                acc[4 * b + 2][v] = fmaf(t, zj, acc[4 * b + 2][v]);
                acc[4 * b + 3][v] += t;
            }
        }

        // ---- rotate pipeline ----
        xj = nx; yj = ny; zj = nz;
        jt = jn;
    }

    // Reduce across the 16 lanes of each half (column dimension of the tile).
    #pragma unroll
    for (int q = 0; q < 4 * NBW; ++q)
        #pragma unroll
        for (int v = 0; v < 8; ++v) {
            float s = acc[q][v];
            s += __shfl_xor(s, 8, 32);
            s += __shfl_xor(s, 4, 32);
            s += __shfl_xor(s, 2, 32);
            s += __shfl_xor(s, 1, 32);
            acc[q][v] = s;
        }

    __shared__ float part[16][8][4 * NBW];
    if (lp == 0) {
        #pragma unroll
        for (int v = 0; v < 8; ++v) {
            const int rl = v + 8 * half;
            #pragma unroll
            for (int q = 0; q < 4 * NBW; ++q) part[rl][wave][q] = acc[q][v];
        }
    }
    __syncthreads();

    if (tid < 16) {
        float tot[4 * NBW];
        #pragma unroll
        for (int q = 0; q < 4 * NBW; ++q) {
            float s = 0.0f;
            #pragma unroll
            for (int wv = 0; wv < 8; ++wv) s += part[tid][wv][q];
            tot[q] = s;
        }
        const int row = i0 + tid;
        const float w0 = w[row * NBW + 0];
        const float w1 = w[row * NBW + 1];
        const float w2 = w[row * NBW + 2];
        const float wb[NBW] = {w0, w1, w2};
        float rx = 0.0f, ry = 0.0f, rz = 0.0f;
        #pragma unroll
        for (int b = 0; b < NBW; ++b) {
            const float s = wb[b] / tot[4 * b + 3];   // w_b / D_b (D_b >= 1: self)
            rx = fmaf(s, tot[4 * b + 0], rx);
            ry = fmaf(s, tot[4 * b + 1], ry);
            rz = fmaf(s, tot[4 * b + 2], rz);
        }
        const float invws = 1.0f / (w0 + w1 + w2);    // faithfulness: softmax row sum
        Xout[3 * row + 0] = rx * invws;
        Xout[3 * row + 1] = ry * invws;
        Xout[3 * row + 2] = rz * invws;
    }
}

// ---------------------------------------------------------------- launch -----

extern "C" void kernel_launch(void* const* d_in, const int* in_sizes, int n_in,
                              void* d_out, int out_size, void* d_ws, size_t ws_size,
                              hipStream_t stream) {
    const float* X     = (const float*)d_in[0];
    const float* Xfea  = (const float*)d_in[1];
    const float* W1    = (const float*)d_in[2];
    const float* b1    = (const float*)d_in[3];
    const float* gamma = (const float*)d_in[4];
    const float* beta  = (const float*)d_in[5];
    const float* W2    = (const float*)d_in[6];
    const float* b2    = (const float*)d_in[7];
    float* out  = (float*)d_out;                 // [4][N][3] history
    float* sums = (float*)d_ws;                  // 256 floats (sum + sumsq)
    float* wbuf = sums + 2 * NITER * DFEA;       // [4][N][3] softmax weights

    zero_stats_kernel<<<1, 256, 0, stream>>>(sums);
    stats_kernel<<<dim3(40, NITER), 256, 0, stream>>>(Xfea, W1, b1, sums);
    weights_kernel<<<dim3(40, NITER), 256, 0, stream>>>(Xfea, W1, b1, gamma, beta,
                                                        W2, b2, sums, wbuf);
    for (int it = 0; it < NITER; ++it) {
        const float* Xin = (it == 0) ? X : (out + (size_t)(it - 1) * NPTS * 3);
        meanshift_kernel<<<NTILE, 256, 0, stream>>>(
            Xin, wbuf + (size_t)it * NPTS * 3, out + (size_t)it * NPTS * 3);
    }
}